// GCNEC_79671643341632
// MI455X (gfx1250) — compile-verified
//
#include <hip/hip_runtime.h>
#include <hip/hip_bf16.h>

typedef float v2f __attribute__((ext_vector_type(2)));
typedef float v8f __attribute__((ext_vector_type(8)));

#define N_NODES 50000
#define N_EDGES 800000

// ---------------- degree accumulation ----------------
__global__ __launch_bounds__(256) void k_deg(const int* __restrict__ src,
                                             const int* __restrict__ dst,
                                             float* __restrict__ deg_out,
                                             float* __restrict__ deg_in, int nE) {
    int e = blockIdx.x * blockDim.x + threadIdx.x;
    if (e < nE) {
        unsafeAtomicAdd(&deg_out[src[e]], 1.0f);
        unsafeAtomicAdd(&deg_in[dst[e]], 1.0f);
    }
}

__global__ __launch_bounds__(256) void k_rsqrt(float* __restrict__ d, int n) {
    int i = blockIdx.x * blockDim.x + threadIdx.x;
    if (i < n) d[i] = rsqrtf(fmaxf(d[i], 1.0f));
}

// ---- edge aggregation: agg[dst] += x[src] * rs_out[src]; d = feature dim (pow2) ----
// Consecutive threads cover consecutive float4 chunks of one edge's feature row,
// so gathers and atomics are coalesced; agg + x stay L2-resident (192 MB L2).
__global__ __launch_bounds__(256) void k_edge_agg(const float* __restrict__ x,
                                                  const float* __restrict__ rs_out,
                                                  const int* __restrict__ src,
                                                  const int* __restrict__ dst,
                                                  float* __restrict__ agg,
                                                  int nE, int d, int lc /* log2(d/4) */) {
    long long tid = (long long)blockIdx.x * blockDim.x + threadIdx.x;
    long long total = (long long)nE << lc;
    if (tid >= total) return;
    int e = (int)(tid >> lc);
    int c = ((int)tid & ((1 << lc) - 1)) << 2;
    int s = src[e], t = dst[e];
    float sc = rs_out[s];
    const float4 v = *reinterpret_cast<const float4*>(x + (size_t)s * d + c);
    float* a = agg + (size_t)t * d + c;
    unsafeAtomicAdd(a + 0, v.x * sc);
    unsafeAtomicAdd(a + 1, v.y * sc);
    unsafeAtomicAdd(a + 2, v.z * sc);
    unsafeAtomicAdd(a + 3, v.w * sc);
}

// ---- C[m][n] = relu( rs[m] * (A[m][:] @ B)[n] + bias[n] ), one wave per 16x16 tile ----
// A: [M,K] row-major, B: [K,N] row-major. K compile-time -> fully unrolled WMMA chain.
// WMMA f32 16x16x4 layouts (ISA 7.12.2):
//   A lane: m = lane&15, holds K = k + (lane>>4)*2 + {0,1}
//   B lane: n = lane&15, holds K rows k + (lane>>4)*2 + {0,1}
//   C VGPR v: row m0 + (lane>>4)*8 + v, col n0 + (lane&15)
template <int K>
__global__ __launch_bounds__(32) void k_gemm_rbr(const float* __restrict__ A,
                                                 const float* __restrict__ rs,
                                                 const float* __restrict__ B,
                                                 const float* __restrict__ bias,
                                                 float* __restrict__ C,
                                                 int N) {
    const int lane = threadIdx.x;
    const int half = lane >> 4;
    const int l16  = lane & 15;
    const int m0 = blockIdx.x << 4;
    const int n0 = blockIdx.y << 4;
    const int mA = m0 + l16;
    const int nB = n0 + l16;
    const float* Arow = A + (size_t)mA * K + (half << 1);
    const float* Bcol = B + nB + (size_t)(half << 1) * N;
    v8f acc = {};
#pragma unroll
    for (int k = 0; k < K; k += 4) {
        v2f a = *reinterpret_cast<const v2f*>(Arow + k);   // straight load -> WMMA
        v2f b;
        b.x = Bcol[(size_t)k * N];
        b.y = Bcol[(size_t)k * N + N];
        acc = __builtin_amdgcn_wmma_f32_16x16x4_f32(false, a, false, b,
                                                    (short)0, acc, false, false);
    }
    // epilogue: per-row scale (factored out of the k-sum), bias, relu
    const float bn = bias[nB];
    const int mrow = m0 + (half << 3);
    float* Crow = C + (size_t)mrow * N + nB;
#pragma unroll
    for (int v = 0; v < 8; ++v) {
        float o = acc[v] * rs[mrow + v] + bn;
        Crow[(size_t)v * N] = fmaxf(o, 0.0f);
    }
}

// ---- pcat[m][0:16] = X@Wp[:128], pcat[m][16:32] = X@Wp[128:]; X:[M,128], Wp:[256,16] ----
__global__ __launch_bounds__(32) void k_proj(const float* __restrict__ X,
                                             const float* __restrict__ Wp,
                                             float* __restrict__ P) {
    const int lane = threadIdx.x;
    const int half = lane >> 4;
    const int l16  = lane & 15;
    const int m0 = blockIdx.x << 4;
    const int n0 = blockIdx.y << 4;          // 0 or 16
    const int cg = n0 + l16;                 // 0..31
    const int mA = m0 + l16;
    const float* Arow = X + (size_t)mA * 128 + (half << 1);
    // col cg<16 -> Wp[d][cg];  cg>=16 -> Wp[128+d][cg-16];  row stride = 16 floats
    const float* Bcol = Wp + ((cg < 16) ? (size_t)cg : (size_t)(128 * 16 + (cg - 16)))
                           + (size_t)(half << 1) * 16;
    v8f acc = {};
#pragma unroll
    for (int k = 0; k < 128; k += 4) {
        v2f a = *reinterpret_cast<const v2f*>(Arow + k);
        v2f b;
        b.x = Bcol[k * 16];
        b.y = Bcol[k * 16 + 16];
        acc = __builtin_amdgcn_wmma_f32_16x16x4_f32(false, a, false, b,
                                                    (short)0, acc, false, false);
    }
    float* Prow = P + (size_t)(m0 + (half << 3)) * 32 + cg;
#pragma unroll
    for (int v = 0; v < 8; ++v) Prow[(size_t)v * 32] = acc[v];
}

// ---- score[e] = pcat[esrc[e]][0:16] + pcat[edst[e]][16:32] + bp ----
__global__ __launch_bounds__(256) void k_score(const float* __restrict__ P,
                                               const int* __restrict__ esrc,
                                               const int* __restrict__ edst,
                                               const float* __restrict__ bp,
                                               float* __restrict__ out, int nE) {
    int e = blockIdx.x * blockDim.x + threadIdx.x;
    if (e >= nE) return;
    int s = esrc[e], t = edst[e];
    const float4* ps = reinterpret_cast<const float4*>(P + (size_t)s * 32);
    const float4* pd = reinterpret_cast<const float4*>(P + (size_t)t * 32 + 16);
    const float4* b4 = reinterpret_cast<const float4*>(bp);
    float4* o = reinterpret_cast<float4*>(out + (size_t)e * 16);
#pragma unroll
    for (int i = 0; i < 4; ++i) {
        float4 a = ps[i], b = pd[i], c = b4[i];
        o[i] = make_float4(a.x + b.x + c.x, a.y + b.y + c.y,
                           a.z + b.z + c.z, a.w + b.w + c.w);
    }
}

extern "C" void kernel_launch(void* const* d_in, const int* in_sizes, int n_in,
                              void* d_out, int out_size, void* d_ws, size_t ws_size,
                              hipStream_t stream) {
    const float* x0   = (const float*)d_in[0];
    const int*   src  = (const int*)d_in[1];
    const int*   dst  = (const int*)d_in[2];
    const int*   esrc = (const int*)d_in[3];
    const int*   edst = (const int*)d_in[4];
    const float* W1   = (const float*)d_in[5];
    const float* b1   = (const float*)d_in[6];
    const float* W2   = (const float*)d_in[7];
    const float* b2   = (const float*)d_in[8];
    const float* Wp   = (const float*)d_in[9];
    const float* bp   = (const float*)d_in[10];
    float* out = (float*)d_out;

    // workspace layout (floats): [rs_out 51200][rs_in 51200][bufA 50000*256][bufB 50000*256]
    float* ws     = (float*)d_ws;
    const size_t PADN = 51200;
    float* rs_out = ws;
    float* rs_in  = ws + PADN;
    float* bufA   = ws + 2 * PADN;
    float* bufB   = bufA + (size_t)N_NODES * 256;

    // 1) degrees -> rsqrt(max(deg,1))
    hipMemsetAsync(rs_out, 0, 2 * PADN * sizeof(float), stream);
    hipMemsetAsync(bufA, 0, (size_t)N_NODES * 128 * sizeof(float), stream);
    k_deg<<<(N_EDGES + 255) / 256, 256, 0, stream>>>(src, dst, rs_out, rs_in, N_EDGES);
    k_rsqrt<<<(int)(2 * PADN + 255) / 256, 256, 0, stream>>>(rs_out, (int)(2 * PADN));

    // 2) agg1[dst] += x0[src] * rs_out[src]   (d=128)
    {
        long long total = (long long)N_EDGES * 32;
        k_edge_agg<<<(int)((total + 255) / 256), 256, 0, stream>>>(
            x0, rs_out, src, dst, bufA, N_EDGES, 128, 5);
    }
    // 3) x1 = relu(rs_in[m] * (agg1 @ W1) + b1)  -> bufB [50000,256]
    k_gemm_rbr<128><<<dim3(N_NODES / 16, 256 / 16), 32, 0, stream>>>(
        bufA, rs_in, W1, b1, bufB, 256);

    // 4) agg2[dst] += x1[src] * rs_out[src]   (d=256)
    hipMemsetAsync(bufA, 0, (size_t)N_NODES * 256 * sizeof(float), stream);
    {
        long long total = (long long)N_EDGES * 64;
        k_edge_agg<<<(int)((total + 255) / 256), 256, 0, stream>>>(
            bufB, rs_out, src, dst, bufA, N_EDGES, 256, 6);
    }
    // 5) x2 = relu(rs_in[m] * (agg2 @ W2) + b2)  -> bufB [50000,128]
    k_gemm_rbr<256><<<dim3(N_NODES / 16, 128 / 16), 32, 0, stream>>>(
        bufA, rs_in, W2, b2, bufB, 128);

    // 6) pcat = [x2 @ Wp_top | x2 @ Wp_bot]   -> bufA [50000,32]
    k_proj<<<dim3(N_NODES / 16, 2), 32, 0, stream>>>(bufB, Wp, bufA);

    // 7) score[e] = pcat[esrc][:16] + pcat[edst][16:] + bp
    k_score<<<(N_EDGES + 255) / 256, 256, 0, stream>>>(bufA, esrc, edst, bp, out, N_EDGES);
}